// GaussianFC_42717744726248
// MI455X (gfx1250) — compile-verified
//
#include <hip/hip_runtime.h>

typedef __attribute__((ext_vector_type(2))) float v2f;
typedef __attribute__((ext_vector_type(8))) float v8f;

#define IN_F   8192
#define OUT_F  8192
#define BATCH  64
#define KC     128              // K-chunk staged in LDS
#define XSTRIDE (KC + 4)        // 132 floats: 16B-aligned rows, conflict-free
#define WAVES  4
#define BLOCK  (WAVES * 32)     // 128 threads
#define COLS_PER_BLOCK (WAVES * 16)  // 64 columns per block

__global__ __launch_bounds__(BLOCK)
void gaussian_fc_wmma(const float* __restrict__ x,
                      const float* __restrict__ mu,
                      const float* __restrict__ sigma,
                      const float* __restrict__ amplitude,
                      const float* __restrict__ bias,
                      float* __restrict__ out)
{
    __shared__ float lds_x[BATCH * XSTRIDE];   // 33,792 B
    __shared__ float lds_mu[KC];
    __shared__ float lds_s[KC];                // -log2(e) / (2*sigma^2)
    __shared__ float lds_amp[KC];

    const int tid  = threadIdx.x;
    const int wave = tid >> 5;
    const int lane = tid & 31;
    const int half = lane >> 4;   // 0: K pair {0,1}; 1: K pair {2,3}
    const int l    = lane & 15;

    const int   n0 = blockIdx.x * COLS_PER_BLOCK + wave * 16;
    const int   n  = n0 + l;
    const float fn = (float)n;

    v8f acc[4] = {};   // 4 M-tiles x 16x16 f32 accumulators

    const float NEG_LOG2E = -1.4426950408889634f;

    for (int k0 = 0; k0 < IN_F; k0 += KC) {
        // ---- stage Gaussian coefficients (pre-fold exp->exp2 scale) ----
        if (tid < KC) {
            float m  = mu[k0 + tid];
            float sg = sigma[k0 + tid];
            float a  = amplitude[k0 + tid];
            lds_mu[tid]  = m;
            lds_s[tid]   = NEG_LOG2E / (2.0f * sg * sg);
            lds_amp[tid] = a;
        }
        // ---- stage x[0:64, k0:k0+KC] into LDS, coalesced float4 ----
        #pragma unroll
        for (int it = 0; it < (BATCH * KC / 4) / BLOCK; ++it) {
            int i4  = tid + it * BLOCK;
            int idx = i4 * 4;
            int m   = idx >> 7;          // idx / KC
            int j   = idx & (KC - 1);    // idx % KC
            float4 v = *(const float4*)(x + (size_t)m * IN_F + k0 + j);
            *(float4*)(&lds_x[m * XSTRIDE + j]) = v;
        }
        __syncthreads();

        // ---- K loop: generate B fragment on the fly, 4 WMMAs reuse it ----
        #pragma unroll 4
        for (int kk = 0; kk < KC; kk += 4) {
            const int kb = kk + 2 * half;   // this half-wave's K pair

            v2f mm = *(const v2f*)(&lds_mu[kb]);
            v2f ss = *(const v2f*)(&lds_s[kb]);
            v2f aa = *(const v2f*)(&lds_amp[kb]);

            float d0 = fn - mm.x;
            float d1 = fn - mm.y;
            v2f b;
            b.x = aa.x * __builtin_amdgcn_exp2f(d0 * d0 * ss.x);
            b.y = aa.y * __builtin_amdgcn_exp2f(d1 * d1 * ss.y);

            #pragma unroll
            for (int mt = 0; mt < 4; ++mt) {
                const int row = mt * 16 + l;
                v2f a = *(const v2f*)(&lds_x[row * XSTRIDE + kb]);
                acc[mt] = __builtin_amdgcn_wmma_f32_16x16x4_f32(
                    /*neg_a=*/false, a, /*neg_b=*/false, b,
                    /*c_mod=*/(short)0, acc[mt],
                    /*reuse_a=*/false, /*reuse_b=*/false);
            }
        }
        __syncthreads();   // protect LDS before next chunk overwrites it
    }

    // ---- epilogue: bias + ReLU, C/D layout per ISA 7.12.2 ----
    const float bv = bias[n];
    #pragma unroll
    for (int mt = 0; mt < 4; ++mt) {
        #pragma unroll
        for (int v = 0; v < 8; ++v) {
            int row = mt * 16 + v + 8 * half;
            float r = acc[mt][v] + bv;
            out[(size_t)row * OUT_F + n] = r > 0.0f ? r : 0.0f;
        }
    }
}

extern "C" void kernel_launch(void* const* d_in, const int* in_sizes, int n_in,
                              void* d_out, int out_size, void* d_ws, size_t ws_size,
                              hipStream_t stream) {
    const float* x         = (const float*)d_in[0];
    const float* mu        = (const float*)d_in[1];
    const float* sigma     = (const float*)d_in[2];
    const float* amplitude = (const float*)d_in[3];
    const float* bias      = (const float*)d_in[4];
    float* out             = (float*)d_out;

    dim3 grid(OUT_F / COLS_PER_BLOCK);   // 128 blocks
    dim3 block(BLOCK);                   // 128 threads = 4 waves
    gaussian_fc_wmma<<<grid, block, 0, stream>>>(x, mu, sigma, amplitude, bias, out);
}